// FlexibleGNNEncoder_28845000360070
// MI455X (gfx1250) — compile-verified
//
#include <hip/hip_runtime.h>
#include <hip/hip_bf16.h>

#define N_NODES 100000
#define N_EDGES 1600000
#define DIM 64
#define N_LAYERS 3
#define N_GRAPHS 128
#define BN_EPS 1e-5f

typedef __attribute__((ext_vector_type(2))) float v2f;
typedef __attribute__((ext_vector_type(8))) float v8f;

// ---------------- degree / normalization ----------------
__global__ void k_init_deg(float* deg) {
  int i = blockIdx.x * blockDim.x + threadIdx.x;
  if (i < N_NODES) deg[i] = 1.0f;  // self-loop contributes 1
}

__global__ void k_count_deg(const int* __restrict__ dst, float* deg) {
  int e = blockIdx.x * blockDim.x + threadIdx.x;
  if (e < N_EDGES) atomicAdd(&deg[dst[e]], 1.0f);
}

__global__ void k_finish_dinv(float* dinv) {
  int i = blockIdx.x * blockDim.x + threadIdx.x;
  if (i < N_NODES) dinv[i] = rsqrtf(dinv[i]);  // deg >= 1 always
}

// ---------------- dense transform t = h @ W via fp32 WMMA ----------------
// One wave per 16x16 output tile. K-loop of 16 x v_wmma_f32_16x16x4_f32.
// A (16x4 f32) layout: lanes 0-15 -> M=0..15 {K=kb,kb+1}; lanes 16-31 -> {K=kb+2,kb+3}.
// B (4x16 f32) layout mirrored over N. C/D: VGPR j holds M=j (lanes 0-15), M=j+8 (lanes 16-31).
__global__ void k_gemm_wmma(const float* __restrict__ h,
                            const float* __restrict__ W,
                            float* __restrict__ t) {
  const int wid  = (blockIdx.x * blockDim.x + threadIdx.x) >> 5;
  const int lane = threadIdx.x & 31;
  const int ntiles = (N_NODES / 16) * (DIM / 16);  // 25000, grid sized exactly
  if (wid >= ntiles) return;                       // wave-uniform (EXEC stays all-1)
  const int tr = wid >> 2;   // row tile
  const int tc = wid & 3;    // col tile (DIM/16 == 4)
  const int row = tr * 16 + (lane & 15);
  const int col = tc * 16 + (lane & 15);
  const int khalf = (lane >> 4) << 1;  // 0 for lanes 0-15, 2 for lanes 16-31

  v8f acc = {};
#pragma unroll
  for (int kb = 0; kb < DIM; kb += 4) {
    v2f a, b;
    a.x = h[(size_t)row * DIM + kb + khalf];
    a.y = h[(size_t)row * DIM + kb + khalf + 1];
    b.x = W[(size_t)(kb + khalf) * DIM + col];
    b.y = W[(size_t)(kb + khalf + 1) * DIM + col];
    acc = __builtin_amdgcn_wmma_f32_16x16x4_f32(false, a, false, b,
                                                (short)0, acc, false, false);
  }

  const int rbase = tr * 16 + ((lane >> 4) << 3);
#pragma unroll
  for (int j = 0; j < 8; ++j)
    t[(size_t)(rbase + j) * DIM + col] = acc[j];
}

// ---------------- aggregation ----------------
// Initialize agg with the self-loop term: agg[i,:] = t[i,:] * dinv[i]^2
__global__ void k_init_agg(const float* __restrict__ t,
                           const float* __restrict__ dinv,
                           float* __restrict__ agg) {
  int tid = blockIdx.x * blockDim.x + threadIdx.x;
  if (tid < N_NODES * DIM) {
    int i = tid >> 6;
    float di = dinv[i];
    agg[tid] = t[tid] * di * di;
  }
}

// Edge scatter: 16 threads per edge, float4 gather + 4 f32 atomics (L2-resident).
__global__ void k_scatter(const int* __restrict__ src, const int* __restrict__ dst,
                          const float* __restrict__ dinv,
                          const float* __restrict__ t, float* __restrict__ agg) {
  int tid = blockIdx.x * blockDim.x + threadIdx.x;
  int e = tid >> 4;
  if (e >= N_EDGES) return;
  int q = (tid & 15) << 2;  // dim base, 16B aligned
  int s = src[e];
  int d = dst[e];
  float nrm = dinv[s] * dinv[d];
  const float4 v = *reinterpret_cast<const float4*>(&t[(size_t)s * DIM + q]);
  float* ap = &agg[(size_t)d * DIM + q];
  atomicAdd(ap + 0, v.x * nrm);
  atomicAdd(ap + 1, v.y * nrm);
  atomicAdd(ap + 2, v.z * nrm);
  atomicAdd(ap + 3, v.w * nrm);
}

// ---------------- batch-norm ----------------
__global__ void k_zero_stats(float* sums, float* sumsq) {
  int i = threadIdx.x;
  if (i < DIM) { sums[i] = 0.0f; sumsq[i] = 0.0f; }
}

// 256 threads = 4 row-lanes x 64 cols; 250 blocks x 400 rows = 100000 rows.
__global__ void k_stats(const float* __restrict__ agg, float* sums, float* sumsq) {
  __shared__ float ssum[4][DIM];
  __shared__ float ssq[4][DIM];
  const int c  = threadIdx.x & 63;
  const int rl = threadIdx.x >> 6;
  const int r0 = blockIdx.x * 400;
  float s = 0.0f, q = 0.0f;
  for (int r = r0 + rl; r < r0 + 400; r += 4) {
    float v = agg[(size_t)r * DIM + c];
    s += v;
    q += v * v;
  }
  ssum[rl][c] = s;
  ssq[rl][c] = q;
  __syncthreads();
  if (rl == 0) {
    s = ssum[0][c] + ssum[1][c] + ssum[2][c] + ssum[3][c];
    q = ssq[0][c] + ssq[1][c] + ssq[2][c] + ssq[3][c];
    atomicAdd(&sums[c], s);
    atomicAdd(&sumsq[c], q);
  }
}

// Per-column bias b cancels exactly under BN mean-subtraction, so it is omitted.
__global__ void k_bn_params(const float* sums, const float* sumsq,
                            const float* __restrict__ gamma,
                            const float* __restrict__ beta,
                            float* scale, float* shift) {
  int c = threadIdx.x;
  if (c < DIM) {
    const float inv_n = 1.0f / (float)N_NODES;
    float mean = sums[c] * inv_n;
    float var  = sumsq[c] * inv_n - mean * mean;
    float sc   = gamma[c] * rsqrtf(var + BN_EPS);
    scale[c] = sc;
    shift[c] = beta[c] - mean * sc;
  }
}

__global__ void k_bn_relu(const float* __restrict__ agg,
                          const float* __restrict__ scale,
                          const float* __restrict__ shift,
                          float* __restrict__ h) {
  int tid = blockIdx.x * blockDim.x + threadIdx.x;
  if (tid < N_NODES * DIM) {
    int c = tid & 63;
    float v = agg[tid] * scale[c] + shift[c];
    h[tid] = fmaxf(v, 0.0f);
  }
}

// ---------------- mean pool (batch is sorted -> run-length accumulate) ----------------
__global__ void k_zero_pool(float* psum, float* pcnt) {
  int i = blockIdx.x * blockDim.x + threadIdx.x;
  if (i < N_GRAPHS * DIM) psum[i] = 0.0f;
  if (i < N_GRAPHS) pcnt[i] = 0.0f;
}

__global__ void k_pool(const float* __restrict__ h, const int* __restrict__ batch,
                       float* psum, float* pcnt) {
  const int c  = threadIdx.x & 63;
  const int rl = threadIdx.x >> 6;
  const int r0 = blockIdx.x * 128;
  const int rend = (r0 + 128 < N_NODES) ? (r0 + 128) : N_NODES;
  float acc = 0.0f, cnt = 0.0f;
  int curg = -1;
  for (int r = r0 + rl; r < rend; r += 4) {
    int g = batch[r];
    if (g != curg) {
      if (curg >= 0) {
        atomicAdd(&psum[curg * DIM + c], acc);
        if (c == 0) atomicAdd(&pcnt[curg], cnt);
      }
      curg = g; acc = 0.0f; cnt = 0.0f;
    }
    acc += h[(size_t)r * DIM + c];
    cnt += 1.0f;
  }
  if (curg >= 0) {
    atomicAdd(&psum[curg * DIM + c], acc);
    if (c == 0) atomicAdd(&pcnt[curg], cnt);
  }
}

__global__ void k_pool_out(const float* __restrict__ psum,
                           const float* __restrict__ pcnt,
                           float* __restrict__ out) {
  int tid = blockIdx.x * blockDim.x + threadIdx.x;
  if (tid < N_GRAPHS * DIM) {
    int g = tid >> 6;
    out[tid] = psum[tid] / fmaxf(pcnt[g], 1.0f);
  }
}

// ---------------- driver ----------------
extern "C" void kernel_launch(void* const* d_in, const int* in_sizes, int n_in,
                              void* d_out, int out_size, void* d_ws, size_t ws_size,
                              hipStream_t stream) {
  (void)in_sizes; (void)n_in; (void)out_size; (void)ws_size;
  const float* x      = (const float*)d_in[0];
  const int*   eidx   = (const int*)d_in[1];   // [2, E] flat
  const int*   batch  = (const int*)d_in[2];
  const float* Ws     = (const float*)d_in[3]; // [3, 64, 64]
  // d_in[4] = bs: cancels exactly under batch-norm mean subtraction
  const float* gammas = (const float*)d_in[5]; // [3, 64]
  const float* betas  = (const float*)d_in[6]; // [3, 64]
  float*       out    = (float*)d_out;

  float* ws   = (float*)d_ws;
  float* H    = ws;                                // N*DIM
  float* T    = H    + (size_t)N_NODES * DIM;      // N*DIM
  float* AGG  = T    + (size_t)N_NODES * DIM;      // N*DIM
  float* DINV = AGG  + (size_t)N_NODES * DIM;      // N
  float* SUMS = DINV + N_NODES;                    // 64
  float* SUMQ = SUMS + DIM;                        // 64
  float* SCL  = SUMQ + DIM;                        // 64
  float* SHF  = SCL  + DIM;                        // 64
  float* PSUM = SHF  + DIM;                        // 128*64
  float* PCNT = PSUM + N_GRAPHS * DIM;             // 128

  const int* src = eidx;
  const int* dst = eidx + N_EDGES;

  const int B = 256;
  // degree -> dinv
  k_init_deg<<<(N_NODES + B - 1) / B, B, 0, stream>>>(DINV);
  k_count_deg<<<(N_EDGES + B - 1) / B, B, 0, stream>>>(dst, DINV);
  k_finish_dinv<<<(N_NODES + B - 1) / B, B, 0, stream>>>(DINV);

  const float* hin = x;
  for (int l = 0; l < N_LAYERS; ++l) {
    const int gemm_blocks = (N_NODES / 16) * (DIM / 16) / (B / 32);  // 3125, exact
    k_gemm_wmma<<<gemm_blocks, B, 0, stream>>>(hin, Ws + (size_t)l * DIM * DIM, T);
    k_init_agg<<<(N_NODES * DIM + B - 1) / B, B, 0, stream>>>(T, DINV, AGG);
    k_scatter<<<(N_EDGES * 16 + B - 1) / B, B, 0, stream>>>(src, dst, DINV, T, AGG);
    k_zero_stats<<<1, DIM, 0, stream>>>(SUMS, SUMQ);
    k_stats<<<250, B, 0, stream>>>(AGG, SUMS, SUMQ);
    k_bn_params<<<1, DIM, 0, stream>>>(SUMS, SUMQ, gammas + l * DIM, betas + l * DIM, SCL, SHF);
    k_bn_relu<<<(N_NODES * DIM + B - 1) / B, B, 0, stream>>>(AGG, SCL, SHF, H);
    hin = H;
  }

  k_zero_pool<<<(N_GRAPHS * DIM + B - 1) / B, B, 0, stream>>>(PSUM, PCNT);
  k_pool<<<(N_NODES + 127) / 128, B, 0, stream>>>(H, batch, PSUM, PCNT);
  k_pool_out<<<(N_GRAPHS * DIM + B - 1) / B, B, 0, stream>>>(PSUM, PCNT, out);
}